// EMAResetQuantizer_5162550690346
// MI455X (gfx1250) — compile-verified
//
#include <hip/hip_runtime.h>
#include <math.h>

typedef __attribute__((ext_vector_type(2))) float v2f;
typedef __attribute__((ext_vector_type(8))) float v8f;

#define KDIM   512
#define NCODE  1024
#define TDIM   2048
#define NBATCH 32
#define NROWS  (NBATCH * TDIM)        // 65536 rows of xf
#define ROWS_PER_BLOCK 64             // CDNA5: 320KB LDS/WGP lets us stage 64 rows (~132KB)
#define LDS_STRIDE 516                // 512 + 4 pad floats: rows land on distinct bank groups

// workspace layout (byte offsets)
#define WS_CTP    0                         // 256*1024 float2 = 2 MB (K-pair-major codebook)
#define WS_C2     (2097152)                 // 1024 floats
#define WS_ACC    (WS_C2 + 4096)            // 1 float commit accumulator
#define WS_COUNTS (WS_ACC + 256)            // 1024 ints
#define WS_IDX    (WS_COUNTS + 4096)        // 65536 ints

// output layout (float offsets): (out, code_idx, commit_loss, perplexity)
#define OUT_IDX    (NBATCH * KDIM * TDIM)   // 33554432
#define OUT_COMMIT (OUT_IDX + NROWS)        // 33619968

__global__ __launch_bounds__(256) void vq_init(int* __restrict__ counts,
                                               float* __restrict__ acc) {
  int t = blockIdx.x * 256 + threadIdx.x;
  if (t < NCODE) counts[t] = 0;
  if (t == 0) acc[0] = 0.0f;
}

// ctp[p][c] = (codebook[c][2p], codebook[c][2p+1]); p in [0,256), c in [0,1024)
__global__ __launch_bounds__(256) void vq_prep_ctp(const float* __restrict__ cb,
                                                   float* __restrict__ ctp) {
  int g = blockIdx.x * 256 + threadIdx.x;   // 262144 threads
  int p = g >> 10, c = g & 1023;
  float a = cb[(size_t)c * KDIM + 2 * p];
  float b = cb[(size_t)c * KDIM + 2 * p + 1];
  size_t o = ((size_t)p * NCODE + c) * 2;
  ctp[o]     = a;
  ctp[o + 1] = b;
}

__global__ __launch_bounds__(256) void vq_prep_c2(const float* __restrict__ cb,
                                                  float* __restrict__ c2) {
  int c = blockIdx.x * 256 + threadIdx.x;   // 1024 threads
  float s = 0.0f;
  for (int k = 0; k < KDIM; ++k) { float v = cb[(size_t)c * KDIM + k]; s = fmaf(v, v, s); }
  c2[c] = s;
}

// Main kernel: per block, 64 rows of xf vs all 1024 codes via f32 WMMA.
// Each wave: 4 column-tile pairs x 4 row tiles; per k-step: 2 B loads + 4 A loads + 8 WMMAs.
__global__ __launch_bounds__(256, 1) void vq_argmin(
    const float* __restrict__ x,      // [32, 512, 2048]
    const float* __restrict__ ctp,    // [256][1024] float2
    const float* __restrict__ c2,     // [1024]
    int*   __restrict__ code_idx,     // [65536]
    int*   __restrict__ counts,       // [1024]
    float* __restrict__ commit_acc)   // [1]
{
  __shared__ float xlds[ROWS_PER_BLOCK * LDS_STRIDE];   // 132096 B (CDNA5 big LDS)
  __shared__ float xsq[ROWS_PER_BLOCK];
  __shared__ float wval[8 * ROWS_PER_BLOCK];            // 2 KB
  __shared__ int   widx[8 * ROWS_PER_BLOCK];            // 2 KB

  const int tid = threadIdx.x;
  const int rowBase = blockIdx.x * ROWS_PER_BLOCK;
  const int n  = rowBase / TDIM;            // 64 | 2048 => single n per block
  const int t0 = rowBase % TDIM;
  const float* xbase = x + (size_t)n * KDIM * TDIM + t0;

  if (tid < ROWS_PER_BLOCK) xsq[tid] = 0.0f;
  __syncthreads();

  // Stage xf rows: xlds[rr*LDS_STRIDE + k] = x[n, k, t0+rr]; accumulate row norms.
  {
    const int rr = tid & 63;                // each thread stays on one row
    const int kb = tid >> 6;                // 0..3
    float s = 0.0f;
    #pragma unroll 8
    for (int i = 0; i < 128; ++i) {
      int k = kb + i * 4;
      float v = xbase[(size_t)k * TDIM + rr];
      xlds[rr * LDS_STRIDE + k] = v;
      s = fmaf(v, v, s);
    }
    atomicAdd(&xsq[rr], s);
  }
  __syncthreads();

  const int lane   = tid & 31;
  const int wv     = tid >> 5;              // 8 waves
  const int lane15 = lane & 15;
  const int khalf  = lane >> 4;             // A/B frag: lanes 0-15 hold K0,K1; 16-31 hold K2,K3
  const float* aptr0 = &xlds[lane15 * LDS_STRIDE + khalf * 2];
  const float* aptr1 = aptr0 + 16 * LDS_STRIDE;
  const float* aptr2 = aptr0 + 32 * LDS_STRIDE;
  const float* aptr3 = aptr0 + 48 * LDS_STRIDE;

  float runv[4][8];
  int   runi[4][8];
  #pragma unroll
  for (int rt = 0; rt < 4; ++rt)
    #pragma unroll
    for (int i = 0; i < 8; ++i) { runv[rt][i] = 3.4e38f; runi[rt][i] = 0; }

  for (int jp = 0; jp < 4; ++jp) {          // each wave sweeps 4 column-tile pairs
    const int col0 = (wv * 8 + jp * 2) * 16 + lane15;   // second tile = col0 + 16
    const float* bptr = ctp + ((size_t)khalf * NCODE + col0) * 2;
    v8f a0c0 = {}, a1c0 = {}, a2c0 = {}, a3c0 = {};
    v8f a0c1 = {}, a1c1 = {}, a2c1 = {}, a3c1 = {};
    #pragma unroll 4
    for (int k0 = 0; k0 < KDIM; k0 += 4) {
      v2f b0 = *(const v2f*)(bptr + (size_t)k0 * NCODE);        // global b64, coalesced
      v2f b1 = *(const v2f*)(bptr + (size_t)k0 * NCODE + 32);   // +16 cols
      v2f a0 = *(const v2f*)(aptr0 + k0);                       // LDS b64 x4 row tiles
      v2f a1 = *(const v2f*)(aptr1 + k0);
      v2f a2 = *(const v2f*)(aptr2 + k0);
      v2f a3 = *(const v2f*)(aptr3 + k0);
      a0c0 = __builtin_amdgcn_wmma_f32_16x16x4_f32(false, a0, false, b0, (short)0, a0c0, false, false);
      a1c0 = __builtin_amdgcn_wmma_f32_16x16x4_f32(false, a1, false, b0, (short)0, a1c0, false, false);
      a2c0 = __builtin_amdgcn_wmma_f32_16x16x4_f32(false, a2, false, b0, (short)0, a2c0, false, false);
      a3c0 = __builtin_amdgcn_wmma_f32_16x16x4_f32(false, a3, false, b0, (short)0, a3c0, false, false);
      a0c1 = __builtin_amdgcn_wmma_f32_16x16x4_f32(false, a0, false, b1, (short)0, a0c1, false, false);
      a1c1 = __builtin_amdgcn_wmma_f32_16x16x4_f32(false, a1, false, b1, (short)0, a1c1, false, false);
      a2c1 = __builtin_amdgcn_wmma_f32_16x16x4_f32(false, a2, false, b1, (short)0, a2c1, false, false);
      a3c1 = __builtin_amdgcn_wmma_f32_16x16x4_f32(false, a3, false, b1, (short)0, a3c1, false, false);
    }
    const float c2v0 = c2[col0];
    const float c2v1 = c2[col0 + 16];
    #pragma unroll
    for (int i = 0; i < 8; ++i) {           // col0 first, then col1: first-min kept on ties
      float d0 = fmaf(-2.0f, a0c0[i], c2v0);
      float d1 = fmaf(-2.0f, a1c0[i], c2v0);
      float d2 = fmaf(-2.0f, a2c0[i], c2v0);
      float d3 = fmaf(-2.0f, a3c0[i], c2v0);
      if (d0 < runv[0][i]) { runv[0][i] = d0; runi[0][i] = col0; }
      if (d1 < runv[1][i]) { runv[1][i] = d1; runi[1][i] = col0; }
      if (d2 < runv[2][i]) { runv[2][i] = d2; runi[2][i] = col0; }
      if (d3 < runv[3][i]) { runv[3][i] = d3; runi[3][i] = col0; }
      float e0 = fmaf(-2.0f, a0c1[i], c2v1);
      float e1 = fmaf(-2.0f, a1c1[i], c2v1);
      float e2 = fmaf(-2.0f, a2c1[i], c2v1);
      float e3 = fmaf(-2.0f, a3c1[i], c2v1);
      if (e0 < runv[0][i]) { runv[0][i] = e0; runi[0][i] = col0 + 16; }
      if (e1 < runv[1][i]) { runv[1][i] = e1; runi[1][i] = col0 + 16; }
      if (e2 < runv[2][i]) { runv[2][i] = e2; runi[2][i] = col0 + 16; }
      if (e3 < runv[3][i]) { runv[3][i] = e3; runi[3][i] = col0 + 16; }
    }
  }

  // Cross-lane min/argmin within each 16-lane half, first-index tie-break.
  #pragma unroll
  for (int m = 8; m >= 1; m >>= 1) {
    #pragma unroll
    for (int rt = 0; rt < 4; ++rt) {
      #pragma unroll
      for (int i = 0; i < 8; ++i) {
        float ov = __shfl_xor(runv[rt][i], m, 32);
        int   oi = __shfl_xor(runi[rt][i], m, 32);
        if (ov < runv[rt][i] || (ov == runv[rt][i] && oi < runi[rt][i])) {
          runv[rt][i] = ov; runi[rt][i] = oi;
        }
      }
    }
  }
  if (lane15 == 0) {                        // lanes 0 (rows m+0..7) and 16 (rows m+8..15)
    #pragma unroll
    for (int rt = 0; rt < 4; ++rt) {
      #pragma unroll
      for (int i = 0; i < 8; ++i) {
        int row = rt * 16 + khalf * 8 + i;
        wval[wv * ROWS_PER_BLOCK + row] = runv[rt][i];
        widx[wv * ROWS_PER_BLOCK + row] = runi[rt][i];
      }
    }
  }
  __syncthreads();

  // Final cross-wave reduce: one thread per row (threads 0..63 = waves 0,1 fully active).
  if (tid < ROWS_PER_BLOCK) {
    const int row = tid;
    float bv = wval[row];
    int   bi = widx[row];
    #pragma unroll
    for (int w2 = 1; w2 < 8; ++w2) {
      float v  = wval[w2 * ROWS_PER_BLOCK + row];
      int   iw = widx[w2 * ROWS_PER_BLOCK + row];
      if (v < bv || (v == bv && iw < bi)) { bv = v; bi = iw; }
    }
    code_idx[rowBase + row] = bi;
    atomicAdd(&counts[bi], 1);
    float lp = xsq[row] + bv;               // ||x||^2 + min(||c||^2 - 2 x.c)
    #pragma unroll
    for (int m = 16; m >= 1; m >>= 1) lp += __shfl_xor(lp, m, 32);
    if ((tid & 31) == 0) atomicAdd(commit_acc, lp);
  }
}

// out[n][w][t] = codebook[idx[n*T+t]][w]; also emit code_idx as float.
__global__ __launch_bounds__(256) void vq_gather(
    const int* __restrict__ code_idx, const float* __restrict__ cb,
    float* __restrict__ out_x, float* __restrict__ out_idx)
{
  size_t o = (size_t)blockIdx.x * 256 + threadIdx.x;     // 33554432 total
  int n   = (int)(o >> 20);                               // 512*2048 = 1<<20
  int rem = (int)(o & ((1u << 20) - 1));
  int w   = rem >> 11;                                    // /2048
  int t   = rem & 2047;
  int idx = code_idx[(n << 11) + t];
  out_x[o] = cb[(size_t)idx * KDIM + w];
  if (w == 0) out_idx[(n << 11) + t] = (float)idx;
}

__global__ __launch_bounds__(1024) void vq_final(const int* __restrict__ counts,
                                                 const float* __restrict__ acc,
                                                 float* __restrict__ out_scalars) {
  __shared__ float red[1024];
  int t = threadIdx.x;
  float prob = (float)counts[t] * (1.0f / (float)NROWS);
  red[t] = prob * logf(prob + 1e-7f);
  __syncthreads();
  for (int s = 512; s > 0; s >>= 1) {
    if (t < s) red[t] += red[t + s];
    __syncthreads();
  }
  if (t == 0) {
    out_scalars[0] = acc[0] / ((float)NROWS * (float)KDIM);  // commit_loss
    out_scalars[1] = expf(-red[0]);                          // perplexity
  }
}

extern "C" void kernel_launch(void* const* d_in, const int* in_sizes, int n_in,
                              void* d_out, int out_size, void* d_ws, size_t ws_size,
                              hipStream_t stream) {
  const float* x  = (const float*)d_in[0];   // [32, 512, 2048]
  const float* cb = (const float*)d_in[1];   // [1024, 512]
  float* out = (float*)d_out;
  char* ws = (char*)d_ws;
  float* ctp    = (float*)(ws + WS_CTP);
  float* c2     = (float*)(ws + WS_C2);
  float* acc    = (float*)(ws + WS_ACC);
  int*   counts = (int*)(ws + WS_COUNTS);
  int*   cidx   = (int*)(ws + WS_IDX);

  vq_init<<<4, 256, 0, stream>>>(counts, acc);
  vq_prep_ctp<<<1024, 256, 0, stream>>>(cb, ctp);
  vq_prep_c2<<<4, 256, 0, stream>>>(cb, c2);
  vq_argmin<<<NROWS / ROWS_PER_BLOCK, 256, 0, stream>>>(x, ctp, c2, cidx, counts, acc);
  vq_gather<<<(NBATCH * KDIM * TDIM) / 256, 256, 0, stream>>>(cidx, cb, out, out + OUT_IDX);
  vq_final<<<1, 1024, 0, stream>>>(counts, acc, out + OUT_COMMIT);
}